// HeteroGCN_30365418783292
// MI455X (gfx1250) — compile-verified
//
#include <hip/hip_runtime.h>

#define N_TX   200000
#define N_ADDR 300000
#define F_IN   128
#define HID    128
#define OUT_F  64
#define NE     600000

typedef __attribute__((ext_vector_type(2))) float v2f;
typedef __attribute__((ext_vector_type(8))) float v8f;

// ---------------- utility kernels ----------------

__global__ void fill_zero(float* p, long n) {
  long i = (long)blockIdx.x * blockDim.x + threadIdx.x;
  if (i < n) p[i] = 0.0f;
}

__global__ void deg_same(const int* __restrict__ ei, float* deg, int E) {
  int e = blockIdx.x * blockDim.x + threadIdx.x;
  if (e < E) atomicAdd(&deg[ei[E + e]], 1.0f);   // in-degree over col
}

__global__ void deg_bi(const int* __restrict__ ei, float* deg_s, float* deg_d, int E) {
  int e = blockIdx.x * blockDim.x + threadIdx.x;
  if (e < E) {
    atomicAdd(&deg_s[ei[e]], 1.0f);              // out-degree over row (source)
    atomicAdd(&deg_d[ei[E + e]], 1.0f);          // in-degree over col (dest)
  }
}

__global__ void dinv_same(float* d, int n) {     // deg -> rsqrt(deg + 1)  (self loop)
  int i = blockIdx.x * blockDim.x + threadIdx.x;
  if (i < n) d[i] = rsqrtf(d[i] + 1.0f);
}

__global__ void dinv_bip(float* d, int n) {      // deg -> deg>0 ? rsqrt(deg) : 0
  int i = blockIdx.x * blockDim.x + threadIdx.x;
  if (i < n) { float v = d[i]; d[i] = (v > 0.0f) ? rsqrtf(v) : 0.0f; }
}

__global__ void init_bias2(float* __restrict__ out, const float* __restrict__ bA,
                           const float* __restrict__ bB, long total, int N) {
  long i = (long)blockIdx.x * blockDim.x + threadIdx.x;
  if (i < total) { int f = (int)(i % N); out[i] = bA[f] + bB[f]; }
}

// ---------------- fp32 WMMA GEMM:  H[M,N] = (relu?)A[M,K=128] @ W[K,N] ----------------
// One wave owns 16 rows x N cols. Exact grid: M % 32 == 0, 2 waves / 64-thread block,
// so EXEC is all-1s for every v_wmma (ISA requirement).
// A-frag (16x4 f32): lanes 0-15 -> M=lane, K={kk,kk+1}; lanes 16-31 -> K={kk+2,kk+3}.
// B-frag (4x16 f32): lanes 0-15 -> N=lane, K={kk,kk+1}; lanes 16-31 -> K={kk+2,kk+3}.
// C/D (16x16 f32):   vgpr v, lanes 0-15 -> M=v, lanes 16-31 -> M=v+8; N=lane%16.
template<int NT, bool RELU, bool SELF>
__global__ void gemm_wmma(const float* __restrict__ A, const float* __restrict__ W,
                          float* __restrict__ H, float* __restrict__ O,
                          const float* __restrict__ dinv, int M) {
  constexpr int N = NT * 16;
  constexpr int K = 128;
  const int lane = threadIdx.x & 31;
  const int wave = threadIdx.x >> 5;
  const long row0 = ((long)blockIdx.x * (blockDim.x >> 5) + wave) * 16;
  const int lrow = lane & 15;   // row (A) / col (B) within tile
  const int kh   = lane >> 4;   // which K-pair this half-wave holds

  v8f acc[NT] = {};
  const float* arow = A + (row0 + lrow) * (long)K;

  for (int kk = 0; kk < K; kk += 4) {
    float a0 = arow[kk + 2 * kh];
    float a1 = arow[kk + 2 * kh + 1];
    if (RELU) { a0 = fmaxf(a0, 0.0f); a1 = fmaxf(a1, 0.0f); }
    v2f a; a.x = a0; a.y = a1;
#pragma unroll
    for (int t = 0; t < NT; ++t) {
      v2f b;
      b.x = W[(kk + 2 * kh) * N + t * 16 + lrow];
      b.y = W[(kk + 2 * kh + 1) * N + t * 16 + lrow];
      acc[t] = __builtin_amdgcn_wmma_f32_16x16x4_f32(
          false, a, false, b, (short)0, acc[t], false, false);
    }
  }

#pragma unroll
  for (int t = 0; t < NT; ++t) {
#pragma unroll
    for (int v = 0; v < 8; ++v) {
      long r = row0 + v + 8 * kh;
      int  c = t * 16 + lrow;
      float val = acc[t][v];
      H[r * (long)N + c] = val;
      if (SELF) {                       // fused self-loop: out += h * dinv^2
        float dv = dinv[r];
        O[r * (long)N + c] += val * dv * dv;
      }
    }
  }
}

// ---------------- edge scatter: out[col] += h[row] * dinv_r[row]*dinv_c[col] ----------------
// One wave per edge; lane owns N/32 contiguous features (float4/float2 gather + fp32 atomics).
template<int N>
__global__ void scatter_edges(const int* __restrict__ ei, const float* __restrict__ h,
                              const float* __restrict__ dr, const float* __restrict__ dc,
                              float* out, int E) {
  const int lane = threadIdx.x & 31;
  const long e = (long)blockIdx.x * (blockDim.x >> 5) + (threadIdx.x >> 5);
  const int row = ei[e];
  const int col = ei[(long)E + e];
  const float norm = dr[row] * dc[col];
  constexpr int FPL = N / 32;
  const float* hp = h + (long)row * N + lane * FPL;
  float* op = out + (long)col * N + lane * FPL;
#pragma unroll
  for (int f = 0; f < FPL; ++f) atomicAdd(op + f, hp[f] * norm);
}

// ---------------- driver ----------------

extern "C" void kernel_launch(void* const* d_in, const int* in_sizes, int n_in,
                              void* d_out, int out_size, void* d_ws, size_t ws_size,
                              hipStream_t stream) {
  const float* x_tx   = (const float*)d_in[0];
  const float* x_addr = (const float*)d_in[1];
  const int*   e_tt   = (const int*)d_in[2];
  const int*   e_aa   = (const int*)d_in[3];
  const int*   e_at   = (const int*)d_in[4];
  const int*   e_ta   = (const int*)d_in[5];
  const float* W1     = (const float*)d_in[6];   // [4,128,128]
  const float* b1     = (const float*)d_in[7];   // [4,128]
  const float* W2     = (const float*)d_in[8];   // [4,128,64]
  const float* b2     = (const float*)d_in[9];   // [4,64]
  float* out = (float*)d_out;

  float* ws = (float*)d_ws;
  float* dinv_tt   = ws; ws += N_TX;
  float* dinv_aa   = ws; ws += N_ADDR;
  float* dinv_at_s = ws; ws += N_ADDR;
  float* dinv_at_d = ws; ws += N_TX;
  float* dinv_ta_s = ws; ws += N_TX;
  float* dinv_ta_d = ws; ws += N_ADDR;
  float* ht   = ws; ws += (long)N_TX * HID;
  float* ha   = ws; ws += (long)N_ADDR * HID;
  float* hbuf = ws;                               // reused per-relation h

  // --- degrees / dinv (shared by both layers) ---
  const long ndeg = 3L * (N_TX + N_ADDR);
  fill_zero<<<(ndeg + 255) / 256, 256, 0, stream>>>(dinv_tt, ndeg);
  deg_same<<<(NE + 255) / 256, 256, 0, stream>>>(e_tt, dinv_tt, NE);
  deg_same<<<(NE + 255) / 256, 256, 0, stream>>>(e_aa, dinv_aa, NE);
  deg_bi  <<<(NE + 255) / 256, 256, 0, stream>>>(e_at, dinv_at_s, dinv_at_d, NE);
  deg_bi  <<<(NE + 255) / 256, 256, 0, stream>>>(e_ta, dinv_ta_s, dinv_ta_d, NE);
  dinv_same<<<(N_TX   + 255) / 256, 256, 0, stream>>>(dinv_tt,   N_TX);
  dinv_same<<<(N_ADDR + 255) / 256, 256, 0, stream>>>(dinv_aa,   N_ADDR);
  dinv_bip <<<(N_ADDR + 255) / 256, 256, 0, stream>>>(dinv_at_s, N_ADDR);
  dinv_bip <<<(N_TX   + 255) / 256, 256, 0, stream>>>(dinv_at_d, N_TX);
  dinv_bip <<<(N_TX   + 255) / 256, 256, 0, stream>>>(dinv_ta_s, N_TX);
  dinv_bip <<<(N_ADDR + 255) / 256, 256, 0, stream>>>(dinv_ta_d, N_ADDR);

  // exact grids: M/16 row-tiles, 2 waves per 64-thread block (200000/32, 300000/32 exact)
  const int GB_TX = N_TX / 32, GB_AD = N_ADDR / 32;
  const int SB = NE / 8;  // scatter: 8 waves per 256-thread block, exact

  // --- layer 1 (N = HID = 128) ---
  init_bias2<<<((long)N_TX   * HID + 255) / 256, 256, 0, stream>>>(ht, b1 + 0 * HID, b1 + 2 * HID, (long)N_TX * HID, HID);
  init_bias2<<<((long)N_ADDR * HID + 255) / 256, 256, 0, stream>>>(ha, b1 + 1 * HID, b1 + 3 * HID, (long)N_ADDR * HID, HID);

  gemm_wmma<8, false, true ><<<GB_TX, 64, 0, stream>>>(x_tx,   W1 + 0 * F_IN * HID, hbuf, ht, dinv_tt, N_TX);
  scatter_edges<128><<<SB, 256, 0, stream>>>(e_tt, hbuf, dinv_tt, dinv_tt, ht, NE);

  gemm_wmma<8, false, false><<<GB_AD, 64, 0, stream>>>(x_addr, W1 + 2 * F_IN * HID, hbuf, nullptr, nullptr, N_ADDR);
  scatter_edges<128><<<SB, 256, 0, stream>>>(e_at, hbuf, dinv_at_s, dinv_at_d, ht, NE);

  gemm_wmma<8, false, true ><<<GB_AD, 64, 0, stream>>>(x_addr, W1 + 1 * F_IN * HID, hbuf, ha, dinv_aa, N_ADDR);
  scatter_edges<128><<<SB, 256, 0, stream>>>(e_aa, hbuf, dinv_aa, dinv_aa, ha, NE);

  gemm_wmma<8, false, false><<<GB_TX, 64, 0, stream>>>(x_tx,   W1 + 3 * F_IN * HID, hbuf, nullptr, nullptr, N_TX);
  scatter_edges<128><<<SB, 256, 0, stream>>>(e_ta, hbuf, dinv_ta_s, dinv_ta_d, ha, NE);

  // --- layer 2 (N = OUT_F = 64), ReLU fused into A-loads ---
  float* ot = out;
  float* oa = out + (long)N_TX * OUT_F;
  init_bias2<<<((long)N_TX   * OUT_F + 255) / 256, 256, 0, stream>>>(ot, b2 + 0 * OUT_F, b2 + 2 * OUT_F, (long)N_TX * OUT_F, OUT_F);
  init_bias2<<<((long)N_ADDR * OUT_F + 255) / 256, 256, 0, stream>>>(oa, b2 + 1 * OUT_F, b2 + 3 * OUT_F, (long)N_ADDR * OUT_F, OUT_F);

  gemm_wmma<4, true, true ><<<GB_TX, 64, 0, stream>>>(ht, W2 + 0 * HID * OUT_F, hbuf, ot, dinv_tt, N_TX);
  scatter_edges<64><<<SB, 256, 0, stream>>>(e_tt, hbuf, dinv_tt, dinv_tt, ot, NE);

  gemm_wmma<4, true, false><<<GB_AD, 64, 0, stream>>>(ha, W2 + 2 * HID * OUT_F, hbuf, nullptr, nullptr, N_ADDR);
  scatter_edges<64><<<SB, 256, 0, stream>>>(e_at, hbuf, dinv_at_s, dinv_at_d, ot, NE);

  gemm_wmma<4, true, true ><<<GB_AD, 64, 0, stream>>>(ha, W2 + 1 * HID * OUT_F, hbuf, oa, dinv_aa, N_ADDR);
  scatter_edges<64><<<SB, 256, 0, stream>>>(e_aa, hbuf, dinv_aa, dinv_aa, oa, NE);

  gemm_wmma<4, true, false><<<GB_TX, 64, 0, stream>>>(ht, W2 + 3 * HID * OUT_F, hbuf, nullptr, nullptr, N_TX);
  scatter_edges<64><<<SB, 256, 0, stream>>>(e_ta, hbuf, dinv_ta_s, dinv_ta_d, oa, NE);
}